// MemoryBank_80633716015726
// MI455X (gfx1250) — compile-verified
//
#include <hip/hip_runtime.h>
#include <hip/hip_bf16.h>

typedef float v2f __attribute__((ext_vector_type(2)));
typedef float v4f __attribute__((ext_vector_type(4)));
typedef float v8f __attribute__((ext_vector_type(8)));

#define D       640
#define NW      5          // ways
#define NS      5          // shots
#define M_ROWS  50000
#define KCH     160        // 640 / 4
#define NTILES  3125       // 50000 / 16
#define NTOPK   8
#define NEG_INF (-3.402823466e38f)

// ws layout (float units)
#define OFF_B       0            // 160*32 v2f = 10240 floats (pre-swizzled B fragments)
#define OFF_SUPSIM  10240        // 32 floats (sup_sim[way*5+m], 25 used)
#define OFF_TKV     10272        // 40 floats  (topk values [way][8])
#define OFF_TKI     10312        // 40 ints    (topk indices [way][8])
#define OFF_SIM     10368        // 5*50000 floats, sim for memory rows [way][row]

// ---------------------------------------------------------------------------
// Kernel 1: build Q (mean of normalized support shots), B fragments, and the
// 25 support-candidate similarities.
// ---------------------------------------------------------------------------
__global__ __launch_bounds__(256)
void mb_setup_kernel(const float* __restrict__ support, float* __restrict__ ws) {
    __shared__ float red[256];
    __shared__ float Qs[NW * D];      // 12.8 KB
    __shared__ float norm25[25];
    const int tid = threadIdx.x;

    // norms of the 25 support vectors, index v = s*5 + w
    for (int v = 0; v < 25; ++v) {
        const float* p = support + v * D;
        float acc = 0.f;
        for (int d = tid; d < D; d += 256) { float x = p[d]; acc = fmaf(x, x, acc); }
        red[tid] = acc; __syncthreads();
        for (int s = 128; s > 0; s >>= 1) { if (tid < s) red[tid] += red[tid + s]; __syncthreads(); }
        if (tid == 0) norm25[v] = fmaxf(sqrtf(red[0]), 1e-12f);
        __syncthreads();
    }

    // Q[w][d] = (1/5) * sum_s support[s,w,d] / norm(s,w)
    for (int i = tid; i < NW * D; i += 256) {
        int w = i / D, d = i - w * D;
        float acc = 0.f;
        for (int s = 0; s < NS; ++s)
            acc += support[(s * NW + w) * D + d] / norm25[s * NW + w];
        Qs[i] = acc * 0.2f;
    }
    __syncthreads();

    // Pre-swizzled B fragments for V_WMMA_F32_16X16X4_F32:
    // lane L: n = L%16 (way col), g = L/16; vgpr0 = Q[n][k0+2g], vgpr1 = Q[n][k0+2g+1]
    v2f* B = (v2f*)(ws + OFF_B);
    for (int i = tid; i < KCH * 32; i += 256) {
        int kc = i >> 5, lane = i & 31, n = lane & 15, g = lane >> 4;
        v2f val; val.x = 0.f; val.y = 0.f;
        if (n < NW) {
            int k0 = kc * 4 + 2 * g;
            val.x = Qs[n * D + k0];
            val.y = Qs[n * D + k0 + 1];
        }
        B[i] = val;
    }

    // sims for the 5 support candidates of each way:
    // sup_sim[w][m] = dot(Q[w], support[m,w]) / norm(m,w)
    for (int v = 0; v < 25; ++v) {
        int m = v / NW, w = v - m * NW;
        const float* p = support + (m * NW + w) * D;
        const float* q = Qs + w * D;
        float acc = 0.f;
        for (int d = tid; d < D; d += 256) acc = fmaf(p[d], q[d], acc);
        red[tid] = acc; __syncthreads();
        for (int s = 128; s > 0; s >>= 1) { if (tid < s) red[tid] += red[tid + s]; __syncthreads(); }
        if (tid == 0) ws[OFF_SUPSIM + w * NW + m] = red[0] / norm25[m * NW + w];
        __syncthreads();
    }
}

// ---------------------------------------------------------------------------
// Kernel 2: streaming similarity. One 16-row memory tile per wave.
// sim[w][row] = dot(Q[w], memory[row]) / max(||memory[row]||, eps)
// via V_WMMA_F32_16X16X4_F32, 160 k-chunks of 4.
// ---------------------------------------------------------------------------
__global__ __launch_bounds__(256)
void mb_sim_kernel(const float* __restrict__ memory, float* __restrict__ ws) {
    __shared__ v2f sB[KCH * 32];   // 40 KB: B fragments shared by all 8 waves
    const int tid = threadIdx.x;
    const v2f* gB = (const v2f*)(ws + OFF_B);
    for (int i = tid; i < KCH * 32; i += 256) sB[i] = gB[i];
    __syncthreads();

    const int wave = tid >> 5, lane = tid & 31;
    const int tile = blockIdx.x * 8 + wave;
    if (tile >= NTILES) return;               // whole-wave uniform exit

    const int row0 = tile * 16;
    const int r = lane & 15, g = lane >> 4;

    // A fragment source: lane holds cols {k0+2g, k0+2g+1} of row (row0+r)
    const v2f* rowA = (const v2f*)(memory + (size_t)(row0 + r) * D + 2 * g);
    const v2f* bp = sB + lane;

    v8f c0 = {}, c1 = {};
    float nrm = 0.f;

    for (int kc = 0; kc < KCH; kc += 2) {
        v2f a0 = rowA[2 * kc];
        v2f a1 = rowA[2 * kc + 2];
        v2f b0 = bp[kc * 32];
        v2f b1 = bp[kc * 32 + 32];
        c0 = __builtin_amdgcn_wmma_f32_16x16x4_f32(false, a0, false, b0,
                                                   (short)0, c0, false, false);
        c1 = __builtin_amdgcn_wmma_f32_16x16x4_f32(false, a1, false, b1,
                                                   (short)0, c1, false, false);
        nrm = fmaf(a0.x, a0.x, nrm); nrm = fmaf(a0.y, a0.y, nrm);
        nrm = fmaf(a1.x, a1.x, nrm); nrm = fmaf(a1.y, a1.y, nrm);
    }
    v8f c = c0 + c1;

    // lane r and lane r+16 each hold half the norm of row (row0+r)
    float nrm2 = nrm + __shfl_xor(nrm, 16);

    // C layout: lane L holds rows 8g..8g+7 (vgpr j -> row 8g+j), way n = L%16.
    // Row-norm of row 8g+j lives in lane 8g+j.
    float inv[8];
    #pragma unroll
    for (int j = 0; j < 8; ++j) {
        float nj = __shfl(nrm2, 8 * g + j);
        inv[j] = 1.0f / fmaxf(sqrtf(nj), 1e-12f);
    }

    const int n = lane & 15;
    if (n < NW) {
        float* simbase = ws + OFF_SIM + (size_t)n * M_ROWS + row0 + 8 * g;
        v4f s0; s0.x = c[0]*inv[0]; s0.y = c[1]*inv[1]; s0.z = c[2]*inv[2]; s0.w = c[3]*inv[3];
        v4f s1; s1.x = c[4]*inv[4]; s1.y = c[5]*inv[5]; s1.z = c[6]*inv[6]; s1.w = c[7]*inv[7];
        *(v4f*)(simbase)     = s0;
        *(v4f*)(simbase + 4) = s1;
    }
}

// ---------------------------------------------------------------------------
// Kernel 3: top-8 per way over 50005 candidates (jax tie-break: lower index).
// ---------------------------------------------------------------------------
__global__ __launch_bounds__(256)
void mb_topk_kernel(float* __restrict__ ws) {
    const int way = blockIdx.x, tid = threadIdx.x;
    __shared__ float lv[2048];
    __shared__ int   li[2048];
    __shared__ float cv[256];
    __shared__ int   ci[256];
    __shared__ int   cs[256];

    float bv[NTOPK]; int bi[NTOPK];
    #pragma unroll
    for (int j = 0; j < NTOPK; ++j) { bv[j] = NEG_INF; bi[j] = 0x7fffffff; }

    const float* supsim = ws + OFF_SUPSIM + way * NW;
    const float* sim    = ws + OFF_SIM + (size_t)way * M_ROWS;

    for (int m = tid; m < M_ROWS + NW; m += 256) {
        float v = (m < NW) ? supsim[m] : sim[m - NW];
        if (v > bv[NTOPK - 1] || (v == bv[NTOPK - 1] && m < bi[NTOPK - 1])) {
            bv[NTOPK - 1] = v; bi[NTOPK - 1] = m;
            #pragma unroll
            for (int j = NTOPK - 1; j > 0; --j) {
                bool up = bv[j] > bv[j-1] || (bv[j] == bv[j-1] && bi[j] < bi[j-1]);
                if (up) {
                    float tv = bv[j]; bv[j] = bv[j-1]; bv[j-1] = tv;
                    int   ti = bi[j]; bi[j] = bi[j-1]; bi[j-1] = ti;
                }
            }
        }
    }
    #pragma unroll
    for (int j = 0; j < NTOPK; ++j) { lv[tid * NTOPK + j] = bv[j]; li[tid * NTOPK + j] = bi[j]; }
    __syncthreads();

    for (int rnd = 0; rnd < NTOPK; ++rnd) {
        // local best among this thread's 8 slots
        float bval = lv[tid * NTOPK]; int bidx = li[tid * NTOPK]; int bslot = tid * NTOPK;
        #pragma unroll
        for (int j = 1; j < NTOPK; ++j) {
            int s = tid * NTOPK + j;
            bool take = lv[s] > bval || (lv[s] == bval && li[s] < bidx);
            if (take) { bval = lv[s]; bidx = li[s]; bslot = s; }
        }
        cv[tid] = bval; ci[tid] = bidx; cs[tid] = bslot;
        __syncthreads();
        for (int s = 128; s > 0; s >>= 1) {
            if (tid < s) {
                bool take = cv[tid+s] > cv[tid] || (cv[tid+s] == cv[tid] && ci[tid+s] < ci[tid]);
                if (take) { cv[tid] = cv[tid+s]; ci[tid] = ci[tid+s]; cs[tid] = cs[tid+s]; }
            }
            __syncthreads();
        }
        if (tid == 0) {
            ws[OFF_TKV + way * NTOPK + rnd] = cv[0];
            ((int*)ws)[OFF_TKI + way * NTOPK + rnd] = ci[0];
            lv[cs[0]] = NEG_INF; li[cs[0]] = 0x7fffffff;
        }
        __syncthreads();
    }
}

// ---------------------------------------------------------------------------
// Kernel 4: weighted prototype: out[w] = sum_i v_i * cand(idx_i) / sum_i v_i
// ---------------------------------------------------------------------------
__global__ __launch_bounds__(256)
void mb_gather_kernel(const float* __restrict__ support,
                      const float* __restrict__ memory,
                      const float* __restrict__ ws,
                      float* __restrict__ out) {
    const int way = blockIdx.x, tid = threadIdx.x;
    __shared__ float sv[NTOPK];
    __shared__ int   si[NTOPK];
    if (tid < NTOPK) {
        sv[tid] = ws[OFF_TKV + way * NTOPK + tid];
        si[tid] = ((const int*)ws)[OFF_TKI + way * NTOPK + tid];
    }
    __syncthreads();

    float den = 0.f;
    #pragma unroll
    for (int i = 0; i < NTOPK; ++i) den += sv[i];

    for (int d = tid; d < D; d += 256) {
        float acc = 0.f;
        #pragma unroll
        for (int i = 0; i < NTOPK; ++i) {
            int m = si[i];
            const float* src = (m < NW) ? (support + (size_t)(m * NW + way) * D)
                                        : (memory + (size_t)(m - NW) * D);
            acc = fmaf(sv[i], src[d], acc);
        }
        out[way * D + d] = acc / den;
    }
}

// ---------------------------------------------------------------------------
extern "C" void kernel_launch(void* const* d_in, const int* in_sizes, int n_in,
                              void* d_out, int out_size, void* d_ws, size_t ws_size,
                              hipStream_t stream) {
    (void)in_sizes; (void)n_in; (void)out_size; (void)ws_size;
    const float* support = (const float*)d_in[0];   // [1,5,5,640]
    const float* memory  = (const float*)d_in[1];   // [50000,640]
    float* out = (float*)d_out;                     // [1,5,640]
    float* ws  = (float*)d_ws;

    mb_setup_kernel<<<1, 256, 0, stream>>>(support, ws);
    mb_sim_kernel<<<(NTILES + 7) / 8, 256, 0, stream>>>(memory, ws);
    mb_topk_kernel<<<NW, 256, 0, stream>>>(ws);
    mb_gather_kernel<<<NW, 256, 0, stream>>>(support, memory, ws, out);
}